// CustomCTC_59957743452403
// MI455X (gfx1250) — compile-verified
//
#include <hip/hip_runtime.h>
#include <hip/hip_bf16.h>
#include <stdint.h>

// Problem constants (match reference)
#define T_   1024
#define B_   128
#define C_   96
#define S_   128
#define L_   257          // 2*S+1
#define TS_  16           // timesteps staged per TDM chunk
#define NCH_ (T_ / TS_)   // 64 chunks
#define NEGF (-1e30f)

typedef __attribute__((ext_vector_type(4))) unsigned int u32x4;
typedef __attribute__((ext_vector_type(8))) int          i32x8;
typedef __attribute__((ext_vector_type(4))) int          i32x4;

#if defined(__AMDGCN__) && __has_builtin(__builtin_amdgcn_tensor_load_to_lds)
#define HAVE_TDM 1
#else
#define HAVE_TDM 0
#endif

#if defined(__AMDGCN__) && __has_builtin(__builtin_amdgcn_s_wait_tensorcnt)
#define WAIT_TENSOR(n) __builtin_amdgcn_s_wait_tensorcnt(n)
#else
#define WAIT_TENSOR(n)
#endif

#if HAVE_TDM
// Issue one TDM 2-D tile load: TS_ rows x C_ floats, row stride B_*C_ elements,
// from global 'gsrc' (tile origin) into LDS at 'lds_dst'.
// D# layout per CDNA5 ISA ch.8 (Tensor DMA Descriptor).
__device__ __forceinline__ void tdm_load_chunk(const float* gsrc, void* lds_dst) {
  unsigned long long ga = (unsigned long long)(uintptr_t)gsrc;
  unsigned lds_off = (unsigned)(uintptr_t)lds_dst;  // low 32 bits of flat LDS addr = LDS byte offset
  // Group 0: [1:0]=count=1, [63:32]=lds_addr, [120:64]=global_addr, [127:126]=type=2
  u32x4 g0 = { 1u,
               lds_off,
               (unsigned)(ga & 0xFFFFFFFFu),
               ((unsigned)((ga >> 32) & 0x01FFFFFFu)) | 0x80000000u };
  // Group 1:
  //  d0: wg_mask=0, data_size=2 (4B) at [17:16]
  //  d1: [15:0] atomic_barrier_addr=0, [31:16] tensor_dim0[15:0]=96
  //  d2: [15:0] tensor_dim0[31:16]=0,  [31:16] tensor_dim1[15:0]=1024
  //  d3: [15:0] tensor_dim1[31:16]=0,  [31:16] tile_dim0=96
  //  d4: [15:0] tile_dim1=TS_,         [31:16] tile_dim2=0
  //  d5: tensor_dim0_stride[31:0] = B_*C_ = 12288 (elements)
  //  d6: [15:0] stride0[47:32]=0, [31:16] stride1[15:0]=0
  //  d7: stride1[47:16]=0
  i32x8 g1 = { (2 << 16),
               (C_ << 16),
               (T_ << 16),
               (C_ << 16),
               TS_,
               (B_ * C_),
               0, 0 };
  i32x4 g2 = { 0, 0, 0, 0 };
  i32x4 g3 = { 0, 0, 0, 0 };
  i32x8 g4 = { 0, 0, 0, 0, 0, 0, 0, 0 };   // 6-arg toolchain variant: extra group
  __builtin_amdgcn_tensor_load_to_lds(g0, g1, g2, g3, g4, 0);
}
#endif

// One workgroup per batch element. 288 threads = 9 waves; threads 0..256 own
// one extended-label state each. Alpha ping-pongs between two LDS arrays with
// a single barrier per timestep; log-prob rows are staged through a
// double-buffered LDS tile filled by the Tensor Data Mover.
__global__ __launch_bounds__(288) void ctc_alpha_kernel(
    const float* __restrict__ log_probs,   // (T, B, C)
    const int*   __restrict__ labels,      // (B, S)
    const int*   __restrict__ input_len,   // (B,)
    const int*   __restrict__ target_len,  // (B,)
    float*       __restrict__ part)        // (B,) per-sample losses
{
  __shared__ float lpb[2][TS_ * C_];   // staged log-prob rows (2 x 6 KB)
  __shared__ float A[2][L_ + 3];       // alpha ping-pong

  const int b = blockIdx.x;
  const int s = threadIdx.x;
  const int ilen_b = input_len[b];

  // Per-thread extended label + skip flag (registers; no LDS needed).
  int ext_s  = 0;   // blank class = 0
  int skip_s = 0;
  if (s < L_ && (s & 1)) {
    const int j = (s - 1) >> 1;
    ext_s = labels[b * S_ + j];
    if (s >= 3) skip_s = (ext_s != labels[b * S_ + j - 1]);
  }

#if HAVE_TDM
  // Prime the pipeline: issue chunk 0 and chunk 1 (wave 0 only; TDM ignores
  // EXEC so the gate must be wave-uniform).
  if (threadIdx.x < 32) {
    tdm_load_chunk(log_probs + (size_t)b * C_,                     &lpb[0][0]);
    tdm_load_chunk(log_probs + ((size_t)TS_ * B_ + b) * C_,        &lpb[1][0]);
  }
#else
  for (int i = threadIdx.x; i < TS_ * C_; i += 288)
    lpb[0][i] = log_probs[((size_t)(i / C_) * B_ + b) * C_ + (i % C_)];
  for (int i = threadIdx.x; i < TS_ * C_; i += 288)
    lpb[1][i] = log_probs[((size_t)(TS_ + i / C_) * B_ + b) * C_ + (i % C_)];
#endif

  for (int ch = 0; ch < NCH_; ++ch) {
#if HAVE_TDM
    if (threadIdx.x < 32) {
      if (ch < NCH_ - 1) { WAIT_TENSOR(1); }   // oldest outstanding (this chunk) done
      else               { WAIT_TENSOR(0); }   // last chunk: nothing may remain
    }
#endif
    __syncthreads();                            // chunk data visible to all waves

    const float* __restrict__ lrow = &lpb[ch & 1][0];

    for (int tl = 0; tl < TS_; ++tl) {
      const int t = ch * TS_ + tl;
      if (s < L_) {
        const float lpe = lrow[tl * C_ + ext_s];   // LDS gather by ext label
        float nv;
        if (t == 0) {
          nv = (s < 2) ? lpe : NEGF;
        } else {
          const float* __restrict__ Ap = A[(t + 1) & 1];  // alpha at t-1
          const float a1 = Ap[s];
          const float a2 = (s >= 1) ? Ap[s - 1] : NEGF;
          const float a3 = skip_s   ? Ap[s - 2] : NEGF;
          const float m  = fmaxf(fmaxf(a1, a2), a3);
          const float sm = __expf(a1 - m) + __expf(a2 - m) + __expf(a3 - m);
          const float cand = m + __logf(sm) + lpe;
          nv = (t < ilen_b) ? cand : a1;           // freeze past input length
        }
        A[t & 1][s] = nv;
      }
      __syncthreads();   // publish alpha[t] before anyone reads it at t+1
    }

#if HAVE_TDM
    // All reads of lpb[ch&1] completed before the barrier above; refill it.
    if (threadIdx.x < 32 && (ch + 2) < NCH_) {
      tdm_load_chunk(log_probs + ((size_t)(ch + 2) * TS_ * B_ + b) * C_,
                     &lpb[ch & 1][0]);
    }
#else
    if ((ch + 2) < NCH_) {
      __syncthreads();
      for (int i = threadIdx.x; i < TS_ * C_; i += 288)
        lpb[ch & 1][i] =
            log_probs[((size_t)((ch + 2) * TS_ + i / C_) * B_ + b) * C_ + (i % C_)];
    }
#endif
  }

  if (threadIdx.x == 0) {
    const float* __restrict__ Af = A[(T_ - 1) & 1];
    const int e = 2 * target_len[b];
    const float ae = Af[e];
    const float al = Af[e - 1];
    const float m  = fmaxf(ae, al);
    const float ll = m + __logf(__expf(ae - m) + __expf(al - m));
    float loss = -ll;
    if (!(loss < 1e20f)) loss = 0.0f;   // zero_infinity (also kills NaN)
    part[b] = loss;
  }
}

// Reduce 128 per-sample losses and apply the focal transform.
__global__ void ctc_reduce_kernel(const float* __restrict__ part,
                                  float* __restrict__ out)
{
  float v = 0.0f;
  for (int k = (int)threadIdx.x; k < B_; k += 32) v += part[k];
  for (int off = 16; off > 0; off >>= 1) v += __shfl_down(v, off, 32);
  if (threadIdx.x == 0) {
    const float ctc = v;
    const float p   = __expf(-ctc);
    float om = 1.0f - p;
    if (om < 0.0f) om = 0.0f;
    out[0] = 0.25f * sqrtf(om) * ctc;   // ALPHA * (1-p)^GAMMA * ctc
  }
}

extern "C" void kernel_launch(void* const* d_in, const int* in_sizes, int n_in,
                              void* d_out, int out_size, void* d_ws, size_t ws_size,
                              hipStream_t stream) {
  const float* log_probs  = (const float*)d_in[0];
  const int*   labels     = (const int*)d_in[1];
  const int*   input_len  = (const int*)d_in[2];
  const int*   target_len = (const int*)d_in[3];
  float*       part       = (float*)d_ws;     // 128 floats of scratch
  float*       out        = (float*)d_out;    // scalar

  ctc_alpha_kernel<<<B_, 288, 0, stream>>>(log_probs, labels, input_len,
                                           target_len, part);
  ctc_reduce_kernel<<<1, 32, 0, stream>>>(part, out);
}